// MessageFunction_45054206935370
// MI455X (gfx1250) — compile-verified
//
#include <hip/hip_runtime.h>

// MI455X / gfx1250 message-function kernel:
//   out[b, 0:128,  n] = relu(W_e @ e_vw[b,:,n] + b_e)
//   out[b, 128:256,n] = relu(W_h @ h_w [b,:,n] + b_h)
// WMMA f32 16x16x4 GEMMs, ping-pong LDS staging with compile-time buffer ids.
// X tile stored K-pair-interleaved in LDS so every B fragment is one aligned
// ds_load_b64; A fragments are contiguous K pairs of W. 128-column block tile
// gives 8-way A-fragment reuse per WMMA operand fetch.

typedef float v2f __attribute__((ext_vector_type(2)));
typedef float v8f __attribute__((ext_vector_type(8)));

#define N_COLS   2048  // N (nodes)
#define K_DIM    256   // feature size (EDGE_F == NODE_F)
#define HALF_M   128   // rows per linear (MSG/2)
#define NTILE    128   // output columns per block
#define KSTEP    16    // K depth staged in LDS per stage (8 k-pairs)
#define WSTRIDE  20    // padded LDS row stride for W tile (words)
#define XSTRIDE2 288   // padded LDS stride per k-pair row of X tile (words; 288%64==32)

__global__ __launch_bounds__(256) void msgfn_wmma_f32(
    const float* __restrict__ h_w,  // [B, 256, 2048]
    const float* __restrict__ e_vw, // [B, 256, 2048]
    const float* __restrict__ W_e,  // [128, 256]
    const float* __restrict__ b_e,  // [128]
    const float* __restrict__ W_h,  // [128, 256]
    const float* __restrict__ b_h,  // [128]
    float* __restrict__ out)        // [B, 256, 2048]
{
    __shared__ float Ws[2][HALF_M * WSTRIDE];        // 128 m-rows x 16 k (padded)
    __shared__ float Xs[2][(KSTEP / 2) * XSTRIDE2];  // 8 k-pair rows x (128 n x 2) (padded)

    const int t    = threadIdx.x;
    const int lane = t & 31;
    const int l15  = t & 15;
    const int hi   = (lane >> 4) & 1;   // upper half of wave32
    const int wave = t >> 5;            // 8 waves
    const int m0   = wave * 16;         // wave's 16-row m tile

    const int n0  = blockIdx.x * NTILE;
    const int src = blockIdx.y;         // 0 = edge linear, 1 = node linear
    const int b   = blockIdx.z;

    const float* Xp = (src == 0 ? e_vw : h_w) + (size_t)b * K_DIM * N_COLS;
    const float* Wp = (src == 0 ? W_e : W_h);
    const float* bp = (src == 0 ? b_e : b_h);

    v8f acc[8];
    #pragma unroll
    for (int c = 0; c < 8; ++c) acc[c] = (v8f){};

    // Cooperative coalesced staging of one K-stage into LDS buffer `buf`.
    auto stage = [&](int buf, int k0) {
        // W tile: 128 rows x 16 k -> Ws[m*20 + k], b128 loads/stores.
        #pragma unroll
        for (int i = 0; i < 2; ++i) {
            const int idx4 = t + i * 256;          // 0..511 float4s
            const int m    = idx4 >> 2;            // 0..127
            const int c4   = (idx4 & 3) << 2;      // 0,4,8,12
            const float4 v = *reinterpret_cast<const float4*>(Wp + m * K_DIM + k0 + c4);
            *reinterpret_cast<float4*>(&Ws[buf][m * WSTRIDE + c4]) = v;
        }
        // X tile: 16 k rows x 128 n, stored pair-interleaved:
        //   Xs[p*XSTRIDE2 + n*2 + (k&1)] = X[k0 + 2p + (k&1)][n0 + n]
        #pragma unroll
        for (int i = 0; i < 2; ++i) {
            const int p  = t >> 5;                 // k-pair row 0..7 (== wave)
            const int n2 = (t & 31) * 2 + i * 64;  // even n 0..126
            const float2 r0 = *reinterpret_cast<const float2*>(
                Xp + (size_t)(k0 + 2 * p) * N_COLS + n0 + n2);
            const float2 r1 = *reinterpret_cast<const float2*>(
                Xp + (size_t)(k0 + 2 * p + 1) * N_COLS + n0 + n2);
            float4 w;                              // {x(k,n), x(k+1,n), x(k,n+1), x(k+1,n+1)}
            w.x = r0.x; w.y = r1.x; w.z = r0.y; w.w = r1.y;
            *reinterpret_cast<float4*>(&Xs[buf][p * XSTRIDE2 + n2 * 2]) = w;
        }
    };

    // One K-stage of compute from LDS buffer `buf`.
    const int arow  = (m0 + l15) * WSTRIDE + 2 * hi;  // A: W[m][k+2hi .. +1]
    const int bbase = hi * XSTRIDE2 + l15 * 2;        // B: X[k+2hi][n], X[k+2hi+1][n]
    auto compute = [&](int buf) {
        const float* Wb = Ws[buf];
        const float* Xb = Xs[buf];
        #pragma unroll
        for (int kk = 0; kk < KSTEP; kk += 4) {
            const v2f a = *reinterpret_cast<const v2f*>(Wb + arow + kk);   // ds_load_b64
            const float* x = Xb + (kk >> 1) * XSTRIDE2 + bbase;
            #pragma unroll
            for (int c = 0; c < 8; ++c) {
                const v2f bf = *reinterpret_cast<const v2f*>(x + c * 32);  // ds_load_b64
                acc[c] = __builtin_amdgcn_wmma_f32_16x16x4_f32(
                    false, a, false, bf, (short)0, acc[c], false, false);
            }
        }
    };

    // Ping-pong pipeline over K = 256 in 16 stages (2 per iteration, buf ids
    // are compile-time constants so accumulators keep one placement).
    stage(0, 0);
    __syncthreads();
    #pragma unroll 1
    for (int s = 0; s < K_DIM / (2 * KSTEP); ++s) {
        const int k0 = s * 2 * KSTEP;
        stage(1, k0 + KSTEP);
        compute(0);
        __syncthreads();
        if (k0 + 2 * KSTEP < K_DIM) stage(0, k0 + 2 * KSTEP);
        compute(1);
        __syncthreads();
    }

    // Epilogue: bias + ReLU, coalesced b32 row stores.
    // C/D layout: VGPR v -> row m0+v (lanes 0-15), m0+v+8 (lanes 16-31); col = n0 + c*16 + l15.
    float* op = out + ((size_t)b * 2 * HALF_M + (size_t)src * HALF_M) * (size_t)N_COLS;
    #pragma unroll
    for (int v = 0; v < 8; ++v) {
        const int m = m0 + v + hi * 8;
        const float bias = bp[m];
        float* row = op + (size_t)m * N_COLS + n0 + l15;
        #pragma unroll
        for (int c = 0; c < 8; ++c) {
            const float r = acc[c][v] + bias;
            row[c * 16] = r > 0.0f ? r : 0.0f;
        }
    }
}

extern "C" void kernel_launch(void* const* d_in, const int* in_sizes, int n_in,
                              void* d_out, int out_size, void* d_ws, size_t ws_size,
                              hipStream_t stream) {
    const float* h_w  = (const float*)d_in[0];
    const float* e_vw = (const float*)d_in[1];
    const float* W_e  = (const float*)d_in[2];
    const float* b_e  = (const float*)d_in[3];
    const float* W_h  = (const float*)d_in[4];
    const float* b_h  = (const float*)d_in[5];
    float* out = (float*)d_out;

    dim3 grid(N_COLS / NTILE, 2, 32);   // (n tiles, edge/node, batch)
    msgfn_wmma_f32<<<grid, 256, 0, stream>>>(h_w, e_vw, W_e, b_e, W_h, b_h, out);

    (void)in_sizes; (void)n_in; (void)out_size; (void)d_ws; (void)ws_size;
}